// SCTCSBLoss_10290741641448
// MI455X (gfx1250) — compile-verified
//
#include <hip/hip_runtime.h>
#include <stdint.h>

#define NF 35            // NUM_FEATURES
#define SLOTS 6          // 32 lanes * 6 = 192 >= L = 2*80+1 = 161
#define RESC 8           // rescale every 8 steps (power of two)
#define LN2 0.6931471805599453f

// ---- gfx1250 async global->LDS path (ASYNCcnt), with safe fallback ----
#if defined(__gfx1250__) && __has_builtin(__builtin_amdgcn_global_load_async_to_lds_b32) && __has_builtin(__builtin_amdgcn_s_wait_asynccnt)
#define USE_ASYNC_LDS 1
typedef __attribute__((address_space(1))) int g_int_t;   // global addrspace
typedef __attribute__((address_space(3))) int l_int_t;   // LDS addrspace
#else
#define USE_ASYNC_LDS 0
#endif

#if __has_builtin(__builtin_amdgcn_rcpf)
#define FAST_RCP(x) __builtin_amdgcn_rcpf(x)
#else
#define FAST_RCP(x) (1.0f / (x))
#endif

// One wave (32 threads) per (b,f) CTC problem. Lane k owns extended-label
// positions [6k, 6k+5]. alpha kept in 6 VGPRs/lane, probability domain with
// periodic power-of-two rescaling (exponent accumulated as an integer).
// Per-t inputs staged in LDS as 16B records -> one ds_load_b128 per step.
extern "C" __global__ void __launch_bounds__(32)
ctc_dp_kernel(const float* __restrict__ logits,   // [T,B,2F+1]
              const int*   __restrict__ targets,  // [B,F,S]
              const int*   __restrict__ in_len,   // [B]
              const int*   __restrict__ tgt_len,  // [B,F]
              float*       __restrict__ loss_out, // [B*F]
              int T, int B, int S) {
    extern __shared__ float4 smem4[];  // (T+1) records: {pos, neg, blank, pad}

    const int prob = blockIdx.x;
    const int b = prob / NF;
    const int f = prob % NF;
    const int lane = threadIdx.x;           // 0..31 (wave32)
    const int C = 2 * NF + 1;
    const size_t tstride = (size_t)B * C;
    const float* gcol = logits + (size_t)b * C;

    // ---- Stage the 3 logit columns for this (b,f) into LDS for all t ----
#if USE_ASYNC_LDS
    for (int t = lane; t < T; t += 32) {
        const float* g = gcol + (size_t)t * tstride;
        float* l = (float*)&smem4[t];
        __builtin_amdgcn_global_load_async_to_lds_b32(
            (g_int_t*)(g + f),      (l_int_t*)(l + 0), 0, 0);
        __builtin_amdgcn_global_load_async_to_lds_b32(
            (g_int_t*)(g + NF + f), (l_int_t*)(l + 1), 0, 0);
        __builtin_amdgcn_global_load_async_to_lds_b32(
            (g_int_t*)(g + 2 * NF), (l_int_t*)(l + 2), 0, 0);
    }
    __builtin_amdgcn_s_wait_asynccnt(0);
#else
    for (int t = lane; t < T; t += 32) {
        const float* g = gcol + (size_t)t * tstride;
        float* l = (float*)&smem4[t];
        l[0] = g[f];
        l[1] = g[NF + f];
        l[2] = g[2 * NF];
    }
#endif
    __syncthreads();

    // ---- Convert staged logits to 3-way softmax probabilities in-place ----
    // (all transcendentals for the whole problem happen here, lane-parallel)
    for (int t = lane; t < T; t += 32) {
        float4 v = smem4[t];
        float m = fmaxf(v.x, fmaxf(v.y, v.z));
        float ex = __expf(v.x - m);
        float ey = __expf(v.y - m);
        float ez = __expf(v.z - m);
        float r = FAST_RCP(ex + ey + ez);
        smem4[t] = make_float4(ex * r, ey * r, ez * r, 0.0f);
    }
    __syncthreads();

    // ---- t-invariant per-slot masks ----
    const int tl    = tgt_len[prob];
    const int inLen = in_len[b];
    const int Lv    = 2 * tl + 1;
    const int* tg   = targets + (size_t)prob * S;

    // Only odd slots (j = 1,3,5) carry labels; even slots are blanks.
    bool validm[SLOTS];
    bool is0m[3], skipm[3];                 // for j = 1,3,5
#pragma unroll
    for (int j = 0; j < SLOTS; ++j) {
        int l = SLOTS * lane + j;
        validm[j] = (l < Lv);
    }
#pragma unroll
    for (int jj = 0; jj < 3; ++jj) {
        int l = SLOTS * lane + 2 * jj + 1;  // odd position
        int k  = (l - 1) >> 1;
        int kk = k < (S - 1) ? k : (S - 1);
        int tv = tg[kk];                    // 0 (pos) or 1 (neg)
        is0m[jj] = (tv == 0);
        bool sk = (l >= 2) && (kk >= 1);
        if (sk) sk = (tv != tg[kk - 1]);    // ext[l] != ext[l-2]
        skipm[jj] = sk;
    }

    // ---- t = 0 initialization (probability domain; invalid = 0) ----
    float a0, a1, a2, a3, a4, a5;
    {
        float4 p0 = smem4[0];
        float e1 = is0m[0] ? p0.x : p0.y;
        a0 = (lane == 0) ? p0.z : 0.0f;
        a1 = (lane == 0 && tl > 0) ? e1 : 0.0f;
        a2 = 0.0f; a3 = 0.0f; a4 = 0.0f; a5 = 0.0f;
    }
    int esum = 0;                           // accumulated log2 of rescales

    const int tend = (T < inLen) ? T : inLen;

    // software-pipelined LDS read (one ds_load_b128 per step, no clamp:
    // record tend (<= T) is allocated and merely unused on the last iteration)
    float4 cur = (tend > 1) ? smem4[1] : make_float4(0.f, 0.f, 0.f, 0.f);

    for (int t = 1; t < tend; ++t) {
        float pp = cur.x, pn = cur.y, pb = cur.z;
        cur = smem4[t + 1];

        float up1 = __shfl_up(a5, 1);       // alpha[l-1] across lane boundary
        if (lane == 0) up1 = 0.0f;

        // fold validity into the per-slot emission factor (0 kills the slot)
        float pb0 = validm[0] ? pb : 0.0f;
        float pb2 = validm[2] ? pb : 0.0f;
        float pb4 = validm[4] ? pb : 0.0f;
        float e1 = validm[1] ? (is0m[0] ? pp : pn) : 0.0f;
        float e3 = validm[3] ? (is0m[1] ? pp : pn) : 0.0f;
        float e5 = validm[5] ? (is0m[2] ? pp : pn) : 0.0f;
        float c1 = skipm[0] ? up1 : 0.0f;
        float c3 = skipm[1] ? a1  : 0.0f;
        float c5 = skipm[2] ? a3  : 0.0f;

        float n0 = (a0 + up1) * pb0;            // even: blank, no skip
        float n2 = (a2 + a1) * pb2;
        float n4 = (a4 + a3) * pb4;
        float n1 = (a1 + a0 + c1) * e1;         // odd: label, optional skip
        float n3 = (a3 + a2 + c3) * e3;
        float n5 = (a5 + a4 + c5) * e5;

        a0 = n0; a1 = n1; a2 = n2; a3 = n3; a4 = n4; a5 = n5;

        if ((t & (RESC - 1)) == 0) {
            // wave-wide max, then multiply by 2^-e and accumulate e
            float mx = fmaxf(fmaxf(fmaxf(a0, a1), fmaxf(a2, a3)), fmaxf(a4, a5));
#pragma unroll
            for (int off = 16; off >= 1; off >>= 1)
                mx = fmaxf(mx, __shfl_xor(mx, off));
            int e = ((__float_as_int(mx) >> 23) & 0xff) - 127;
            float s = __int_as_float((127 - e) << 23);  // 2^-e
            a0 *= s; a1 *= s; a2 *= s; a3 *= s; a4 *= s; a5 *= s;
            esum += e;
        }
    }

    // ---- Readout: ll = log(alpha[2*tl] + alpha[2*tl-1]) + esum*ln2 ----
    float av[SLOTS] = {a0, a1, a2, a3, a4, a5};
    int idx = 2 * tl;
    int r = idx % SLOTS;
    float v = av[0];
#pragma unroll
    for (int j = 1; j < SLOTS; ++j) v = (r == j) ? av[j] : v;
    float last = __shfl(v, idx / SLOTS);

    float prevv = 0.0f;
    if (tl > 0) {                          // uniform branch
        int idxp = idx - 1;
        int r2 = idxp % SLOTS;
        float w = av[0];
#pragma unroll
        for (int j = 1; j < SLOTS; ++j) w = (r2 == j) ? av[j] : w;
        prevv = __shfl(w, idxp / SLOTS);
    }

    float sum = last + prevv;
    float ll = (__log2f(sum) + (float)esum) * LN2;   // natural-log likelihood
    float loss = (ll > -1e29f) ? -ll : 0.0f;         // -inf (sum==0) -> 0

    if (lane == 0) {
        float tlc = (float)(tl > 0 ? tl : 1);
        loss_out[prob] = loss / (tlc * (float)B);    // fold mean-over-B + /tl
    }
}

// Deterministic fixed-order reduction of the B*F per-problem losses.
extern "C" __global__ void __launch_bounds__(256)
ctc_reduce_kernel(const float* __restrict__ loss, float* __restrict__ out, int n) {
    __shared__ float sbuf[256];
    float s = 0.0f;
    for (int i = threadIdx.x; i < n; i += 256) s += loss[i];
    sbuf[threadIdx.x] = s;
    __syncthreads();
    for (int off = 128; off >= 1; off >>= 1) {
        if ((int)threadIdx.x < off) sbuf[threadIdx.x] += sbuf[threadIdx.x + off];
        __syncthreads();
    }
    if (threadIdx.x == 0) out[0] = sbuf[0];
}

extern "C" void kernel_launch(void* const* d_in, const int* in_sizes, int n_in,
                              void* d_out, int out_size, void* d_ws, size_t ws_size,
                              hipStream_t stream) {
    const float* logits  = (const float*)d_in[0];
    const int*   targets = (const int*)d_in[1];
    const int*   in_len  = (const int*)d_in[2];
    const int*   tgt_len = (const int*)d_in[3];

    const int B = in_sizes[2];                 // input_lengths: [B]
    const int C = 2 * NF + 1;
    const int T = in_sizes[0] / (B * C);       // logits: [T,B,2F+1]
    const int S = in_sizes[1] / (B * NF);      // targets: [B,F,S]
    const int nprob = B * NF;

    float* loss_ws = (float*)d_ws;             // nprob floats of scratch
    const size_t shmem = (size_t)(T + 1) * sizeof(float4);

    ctc_dp_kernel<<<nprob, 32, shmem, stream>>>(logits, targets, in_len, tgt_len,
                                                loss_ws, T, B, S);
    ctc_reduce_kernel<<<1, 256, 0, stream>>>(loss_ws, (float*)d_out, nprob);
}